// KAConv_78056735637896
// MI455X (gfx1250) — compile-verified
//
#include <hip/hip_runtime.h>
#include <hip/hip_bf16.h>

typedef __attribute__((ext_vector_type(2))) float v2f;
typedef __attribute__((ext_vector_type(8))) float v8f;

namespace {
constexpr int Bsz  = 4;
constexpr int Cin  = 16;
constexpr int Fout = 16;
constexpr int Hdim = 64;
constexpr int Wdim = 64;
constexpr int NA = Fout * Cin * 9 * 6;   // 13824 numerator coeffs
constexpr int NQ = Fout * Cin * 9 * 4;   //  9216 denominator coeffs
constexpr int SLAB_STRIDE = 68;          // 1 pad + 64 + 1 pad + 2 align
constexpr int SLAB_ROWS   = 6;           // h-1 .. h+4 for the 4 rows this block owns
constexpr int NX = Cin * SLAB_ROWS * SLAB_STRIDE;  // 6528 floats
}

// ---- CDNA5 async-to-LDS copies (ASYNCcnt-tracked; see CDNA5 ISA) ----
__device__ __forceinline__ unsigned lds_off(const void* p) {
    // generic shared pointer: low 32 bits are the wave-relative LDS byte address
    return (unsigned)(unsigned long long)(uintptr_t)p;
}
__device__ __forceinline__ void async_b32(unsigned dst_lds, const void* g) {
    asm volatile("global_load_async_to_lds_b32 %0, %1, off"
                 :: "v"(dst_lds), "v"((unsigned long long)(uintptr_t)g) : "memory");
}
__device__ __forceinline__ void async_b128(unsigned dst_lds, const void* g) {
    asm volatile("global_load_async_to_lds_b128 %0, %1, off"
                 :: "v"(dst_lds), "v"((unsigned long long)(uintptr_t)g) : "memory");
}
__device__ __forceinline__ void wait_async0() {
    asm volatile("s_wait_asynccnt 0x0" ::: "memory");
}

// One wave32 owns TWO adjacent 16(out-channel) x 16(pixel) WMMA tiles (32 pixels),
// sharing the per-(c,p) A-coefficient fragments across both tiles.
// Per (c, p): P via 2 chained V_WMMA_F32_16X16X4_F32 per tile, Q via 1 per tile
// (6 WMMAs), then acc += P * rcp(1+|Q|) elementwise on the accumulator fragments.
__launch_bounds__(256)
__global__ void kaconv_rational_wmma(const float* __restrict__ x,
                                     const float* __restrict__ A,
                                     const float* __restrict__ Bc,
                                     float* __restrict__ out)
{
    extern __shared__ float smem[];
    float* sA = smem;             // [F][C][9][6]
    float* sQ = smem + NA;        // [F][C][9][4]
    float* sX = smem + NA + NQ;   // [C][6 rows][68 cols], col 0 == image col -1

    const int b  = blockIdx.x >> 4;          // 16 blocks per batch image
    const int hb = (blockIdx.x & 15) * 4;    // this block owns rows hb .. hb+3

    // --- stage coefficient tables via async b128 transfers ---
    for (int i = threadIdx.x; i < NA / 4; i += blockDim.x)
        async_b128(lds_off(&sA[i * 4]), A + i * 4);
    for (int i = threadIdx.x; i < NQ / 4; i += blockDim.x)
        async_b128(lds_off(&sQ[i * 4]), Bc + i * 4);

    // --- stage x halo slab: interior via async b32, padding as LDS zeros ---
    for (int i = threadIdx.x; i < Cin * SLAB_ROWS * Wdim; i += blockDim.x) {
        const int c   = i / (SLAB_ROWS * Wdim);
        const int rr  = (i / Wdim) % SLAB_ROWS;  // slab row
        const int col = i & 63;                  // image column
        const int hh  = hb - 1 + rr;             // image row
        float* cell = &sX[(c * SLAB_ROWS + rr) * SLAB_STRIDE + col + 1];
        if (hh >= 0 && hh < Hdim)
            async_b32(lds_off(cell), &x[((size_t)(b * Cin + c) * Hdim + hh) * Wdim + col]);
        else
            *cell = 0.0f;                        // rows outside the image
    }
    for (int i = threadIdx.x; i < Cin * SLAB_ROWS * 4; i += blockDim.x) {
        const int c  = i >> 4;
        const int rr = (i >> 2) & 3;             // covers 4 of 6 rows; next stride covers rest
        // map i over all (c, rr, pad) combos properly:
        // reindex: total = 16*6*4 = 384 -> do plain decomposition instead
        (void)c; (void)rr;
    }
    // pad columns {0, 65, 66, 67} for all (c, row)
    for (int i = threadIdx.x; i < Cin * SLAB_ROWS * 4; i += blockDim.x) {
        const int c   = i / (SLAB_ROWS * 4);
        const int rr  = (i >> 2) % SLAB_ROWS;
        const int pc  = i & 3;
        const int col = (pc == 0) ? 0 : 64 + pc;
        sX[(c * SLAB_ROWS + rr) * SLAB_STRIDE + col] = 0.0f;
    }

    wait_async0();
    __syncthreads();

    const int lane = threadIdx.x & 31;
    const int n    = lane & 15;          // pixel column within tile / A-row f
    const int half = lane >> 4;          // K-half selector for A/B fragments
    const int wid  = threadIdx.x >> 5;
    const int w0   = (wid & 1) << 5;     // 2 tile-pairs per row: cols [w0, w0+32)
    const int hl   = (wid >> 1) & 3;     // which of the block's 4 rows
    const int h    = hb + hl;

    v8f acc0 = {0.f, 0.f, 0.f, 0.f, 0.f, 0.f, 0.f, 0.f};
    v8f acc1 = {0.f, 0.f, 0.f, 0.f, 0.f, 0.f, 0.f, 0.f};

    for (int c = 0; c < Cin; ++c) {
        // per-lane base into the slab: row hl, image col (w0+n), +1 halo shift
        const float* xRow  = &sX[(c * SLAB_ROWS + hl) * SLAB_STRIDE + w0 + n];
        const float* aBase = sA + ((n * Cin + c) * 9) * 6;
        const float* qBase = sQ + ((n * Cin + c) * 9) * 4;
        #pragma unroll
        for (int p = 0; p < 9; ++p) {
            const int di = p / 3, dj = p % 3;
            const float xv0 = xRow[di * SLAB_STRIDE + dj];        // tile 0
            const float xv1 = xRow[di * SLAB_STRIDE + dj + 16];   // tile 1

            // A fragments (16 x K): lane row f = n, half selects K pair.
            // All loads unconditional (addresses always valid) -> cndmask, no exec branch.
            const float* a  = aBase + p * 6;
            const float* q  = qBase + p * 4;
            const int    k0 = half << 1;
            const float  a4 = a[4];
            const float  a5 = a[5];
            v2f aP0, aP1, aQ;
            aP0.x = a[k0];
            aP0.y = a[k0 + 1];
            aP1.x = half ? 0.0f : a4;
            aP1.y = half ? 0.0f : a5;
            aQ.x  = q[k0];
            aQ.y  = q[k0 + 1];

            v8f Pm0 = {0.f,0.f,0.f,0.f,0.f,0.f,0.f,0.f};
            v8f Pm1 = {0.f,0.f,0.f,0.f,0.f,0.f,0.f,0.f};
            v8f Qm0 = {0.f,0.f,0.f,0.f,0.f,0.f,0.f,0.f};
            v8f Qm1 = {0.f,0.f,0.f,0.f,0.f,0.f,0.f,0.f};

            {   // tile 0 B fragments and WMMAs
                const float x2 = xv0 * xv0, x3 = x2 * xv0, x4 = x2 * x2, x5 = x4 * xv0;
                v2f bP0, bP1, bQ;
                bP0.x = half ? x2   : 1.0f;   // K: 0 | 2
                bP0.y = half ? x3   : xv0;    // K: 1 | 3
                bP1.x = half ? 0.0f : x4;     // K: 4 | 6(pad)
                bP1.y = half ? 0.0f : x5;     // K: 5 | 7(pad)
                bQ.x  = half ? x3   : xv0;    // powers x^1..x^4
                bQ.y  = half ? x4   : x2;
                Pm0 = __builtin_amdgcn_wmma_f32_16x16x4_f32(false, aP0, false, bP0, (short)0, Pm0, false, false);
                Pm0 = __builtin_amdgcn_wmma_f32_16x16x4_f32(false, aP1, false, bP1, (short)0, Pm0, false, false);
                Qm0 = __builtin_amdgcn_wmma_f32_16x16x4_f32(false, aQ,  false, bQ,  (short)0, Qm0, false, false);
            }
            {   // tile 1 B fragments and WMMAs (A fragments reused)
                const float x2 = xv1 * xv1, x3 = x2 * xv1, x4 = x2 * x2, x5 = x4 * xv1;
                v2f bP0, bP1, bQ;
                bP0.x = half ? x2   : 1.0f;
                bP0.y = half ? x3   : xv1;
                bP1.x = half ? 0.0f : x4;
                bP1.y = half ? 0.0f : x5;
                bQ.x  = half ? x3   : xv1;
                bQ.y  = half ? x4   : x2;
                Pm1 = __builtin_amdgcn_wmma_f32_16x16x4_f32(false, aP0, false, bP0, (short)0, Pm1, false, false);
                Pm1 = __builtin_amdgcn_wmma_f32_16x16x4_f32(false, aP1, false, bP1, (short)0, Pm1, false, false);
                Qm1 = __builtin_amdgcn_wmma_f32_16x16x4_f32(false, aQ,  false, bQ,  (short)0, Qm1, false, false);
            }

            #pragma unroll
            for (int e = 0; e < 8; ++e) {
                const float d0 = 1.0f + __builtin_fabsf(Qm0[e]);
                const float d1 = 1.0f + __builtin_fabsf(Qm1[e]);
                acc0[e] += Pm0[e] * __builtin_amdgcn_rcpf(d0);
                acc1[e] += Pm1[e] * __builtin_amdgcn_rcpf(d1);
            }
        }
    }

    // D fragment: VGPR e -> out channel f = e + 8*half, lane&15 -> pixel col
    float* op = out + (((size_t)b * Fout) * Hdim + h) * Wdim + (w0 + n);
    #pragma unroll
    for (int e = 0; e < 8; ++e) {
        const int f = e + (half << 3);
        op[(size_t)f * (Hdim * Wdim)]      = acc0[e];
        op[(size_t)f * (Hdim * Wdim) + 16] = acc1[e];
    }
}

extern "C" void kernel_launch(void* const* d_in, const int* in_sizes, int n_in,
                              void* d_out, int out_size, void* d_ws, size_t ws_size,
                              hipStream_t stream) {
    const float* x  = (const float*)d_in[0];   // [4,16,64,64]
    const float* A  = (const float*)d_in[1];   // [16,16,9,6]
    const float* Bc = (const float*)d_in[2];   // [16,16,9,4]
    float*       o  = (float*)d_out;           // [4,16,64,64]

    constexpr int totalWaves = Bsz * Hdim * (Wdim / 32);   // 512 (2 tiles/wave)
    dim3 block(256);                                        // 8 waves/block
    dim3 grid(totalWaves / 8);                              // 64 blocks
    size_t shmem = (size_t)(NA + NQ + NX) * sizeof(float);  // ~118 KB (CDNA5: 320KB/WGP)

    hipLaunchKernelGGL(kaconv_rational_wmma, grid, block, shmem, stream,
                       x, A, Bc, o);
}